// SharedEncoderGraph_44092134261233
// MI455X (gfx1250) — compile-verified
//
#include <hip/hip_runtime.h>
#include <math.h>

// CDNA5 / gfx1250 implementation of SharedEncoderGraph forward.
// Pipeline (all on `stream`):
//   1) zero agg + pool scratch
//   2) h = relu(x @ W_in + b_in)                      [WMMA f32 16x16x4]
//   3) agg[src] += edge_val * h[dst]                  [warp/edge, f32 atomics @ L2]
//   4) h_struct = l2norm(relu(agg @ W_gcn))  -> d_out [WMMA + fused norm]
//   5) pool[g]  += batch_val * h_struct               [warp/node, atomics]
//   6) h_graph  = l2norm(relu(pool @ W_g + b_g)) -> d_out + N*128

typedef __attribute__((ext_vector_type(2))) float v2f;
typedef __attribute__((ext_vector_type(8))) float v8f;

#define DEMB 128

// Sum across the 16 lanes of each half-wave (wave32: masks < 16 stay in-half).
__device__ __forceinline__ float half_group_sum16(float v) {
    v += __shfl_xor(v, 1, 32);
    v += __shfl_xor(v, 2, 32);
    v += __shfl_xor(v, 4, 32);
    v += __shfl_xor(v, 8, 32);
    return v;
}

// out[r, :] = epilogue(in[r, :] @ W + bias), W is 128x128, row-major.
// Block = 256 threads = 8 waves; each wave computes 16 rows x 128 cols.
// W is staged transposed in LDS: sWt[n*128 + k] = W[k*128 + n] so each B
// fragment pair lowers to ds_load_2addr_stride64_b64 (2 fragments / instr).
template <bool HAS_BIAS, bool L2NORM>
__global__ __launch_bounds__(256) void gemm128_relu_kernel(
    const float* __restrict__ in, const float* __restrict__ W,
    const float* __restrict__ bias, float* __restrict__ out, int nrows) {
    __shared__ float sWt[DEMB * DEMB];  // 64 KB of the 320 KB WGP LDS
    for (int i = threadIdx.x * 4; i < DEMB * DEMB; i += blockDim.x * 4) {
        const float4 w4 = *(const float4*)&W[i];  // row k, cols n..n+3
        const int k = i >> 7, n = i & 127;
        sWt[(n + 0) * DEMB + k] = w4.x;
        sWt[(n + 1) * DEMB + k] = w4.y;
        sWt[(n + 2) * DEMB + k] = w4.z;
        sWt[(n + 3) * DEMB + k] = w4.w;
    }
    __syncthreads();

    const int wave = threadIdx.x >> 5;
    const int lane = threadIdx.x & 31;
    const int n    = lane & 15;   // column-in-tile / A row select
    const int kg   = lane >> 4;   // half-wave: K group for A/B, +8 rows for C/D
    const int rowBase = blockIdx.x * 128 + wave * 16;

    // A fragment source: this lane feeds row (rowBase + n), elements k, k+1.
    // Clamp to a valid row so the load is always legal; zero via cndmask below
    // (branch-free inner loop; only the final block has invalid rows).
    const int  arowIdx  = rowBase + n;
    const bool rowValid = arowIdx < nrows;
    const int  arowSafe = rowValid ? arowIdx : (nrows - 1);
    const float* __restrict__ arow = in + (size_t)arowSafe * DEMB;
    const float amask = rowValid ? 1.0f : 0.0f;

    v8f acc[8] = {};  // 8 column tiles of 16 -> full 128-wide row stripe

    for (int kb = 0; kb < DEMB; kb += 4) {
        const int k = kb + 2 * kg;
        v2f a = *(const v2f*)(arow + k);
        a.x *= amask;   // v_cndmask-equivalent, no exec manipulation
        a.y *= amask;
#pragma unroll
        for (int t = 0; t < 8; ++t) {
            const v2f b = *(const v2f*)&sWt[(t * 16 + n) * DEMB + k];
            acc[t] = __builtin_amdgcn_wmma_f32_16x16x4_f32(
                false, a, false, b, (short)0, acc[t], false, false);
        }
    }

    // Epilogue: bias + relu in-place.  Lane element acc[t][r] is
    // out[rowBase + r + 8*kg][t*16 + n].
#pragma unroll
    for (int t = 0; t < 8; ++t) {
        const float bv = HAS_BIAS ? bias[t * 16 + n] : 0.0f;
#pragma unroll
        for (int r = 0; r < 8; ++r) {
            acc[t][r] = fmaxf(acc[t][r] + bv, 0.0f);
        }
    }

    if (L2NORM) {
#pragma unroll
        for (int r = 0; r < 8; ++r) {
            float ss = 0.0f;
#pragma unroll
            for (int t = 0; t < 8; ++t) ss += acc[t][r] * acc[t][r];
            ss = half_group_sum16(ss);  // full row sum-of-squares
            const float scale = 1.0f / fmaxf(sqrtf(ss), 1e-12f);
#pragma unroll
            for (int t = 0; t < 8; ++t) acc[t][r] *= scale;
        }
    }

#pragma unroll
    for (int r = 0; r < 8; ++r) {
        const int row = rowBase + r + 8 * kg;
        if (row < nrows) {
#pragma unroll
            for (int t = 0; t < 8; ++t)
                out[(size_t)row * DEMB + t * 16 + n] = acc[t][r];
        }
    }
}

// One wave per edge: agg[src] += val * h[dst]; float4 per lane (128 cols).
// Edge id is forced wave-uniform with readfirstlane so src/dst/val become
// scalar (SMEM) loads instead of 32 redundant vector loads.
__global__ __launch_bounds__(256) void spmm_edge_kernel(
    const int* __restrict__ esrc, const int* __restrict__ edst,
    const float* __restrict__ eval, const float* __restrict__ h,
    float* __restrict__ agg, int E) {
    const int lane = threadIdx.x & 31;
    const long long flat = ((long long)blockIdx.x * blockDim.x) + threadIdx.x;
    const int e = __builtin_amdgcn_readfirstlane((int)(flat >> 5));
    if (e >= E) return;
    const int   s = esrc[e];
    const int   d = edst[e];
    const float v = eval[e];
    const float4 hv = *(const float4*)&h[(size_t)d * DEMB + lane * 4];
    float* ap = &agg[(size_t)s * DEMB + lane * 4];
    atomicAdd(ap + 0, v * hv.x);
    atomicAdd(ap + 1, v * hv.y);
    atomicAdd(ap + 2, v * hv.z);
    atomicAdd(ap + 3, v * hv.w);
}

// One wave per node: pool[batch_ids[i]] += batch_val[i] * h_struct[i].
__global__ __launch_bounds__(256) void pool_node_kernel(
    const float* __restrict__ hs, const float* __restrict__ bval,
    const int* __restrict__ bids, float* __restrict__ pool, int N) {
    const int lane = threadIdx.x & 31;
    const long long flat = ((long long)blockIdx.x * blockDim.x) + threadIdx.x;
    const int i = __builtin_amdgcn_readfirstlane((int)(flat >> 5));
    if (i >= N) return;
    const int   g = bids[i];
    const float w = bval[i];
    const float4 hv = *(const float4*)&hs[(size_t)i * DEMB + lane * 4];
    float* pp = &pool[(size_t)g * DEMB + lane * 4];
    atomicAdd(pp + 0, w * hv.x);
    atomicAdd(pp + 1, w * hv.y);
    atomicAdd(pp + 2, w * hv.z);
    atomicAdd(pp + 3, w * hv.w);
}

__global__ __launch_bounds__(256) void zero_kernel(float* __restrict__ p, size_t n) {
    const size_t i = (size_t)blockIdx.x * blockDim.x + threadIdx.x;
    if (i < n) p[i] = 0.0f;
}

extern "C" void kernel_launch(void* const* d_in, const int* in_sizes, int n_in,
                              void* d_out, int out_size, void* d_ws, size_t ws_size,
                              hipStream_t stream) {
    const float* x        = (const float*)d_in[0];
    const int*   edge_src = (const int*)d_in[1];
    const int*   edge_dst = (const int*)d_in[2];
    const float* edge_val = (const float*)d_in[3];
    const int*   batch_id = (const int*)d_in[4];
    const float* batch_vl = (const float*)d_in[5];
    const float* W_in     = (const float*)d_in[6];
    const float* b_in     = (const float*)d_in[7];
    const float* W_gcn    = (const float*)d_in[8];
    const float* W_g      = (const float*)d_in[9];
    const float* b_g      = (const float*)d_in[10];

    const int N = in_sizes[4];          // batch_ids: one per node
    const int E = in_sizes[1];          // edge_src
    const int G = 1024;                 // num_graphs (reference constant)

    float* out_f    = (float*)d_out;
    float* h_struct = out_f;                         // [N,128] (output 0)
    float* h_graph  = out_f + (size_t)N * DEMB;      // [G,128] (output 1)

    float* h    = (float*)d_ws;                      // [N,128]
    float* agg  = h + (size_t)N * DEMB;              // [N,128]
    float* pool = agg + (size_t)N * DEMB;            // [G,128]

    // 1) zero the accumulation scratch (agg and pool are contiguous).
    {
        const size_t nz = (size_t)N * DEMB + (size_t)G * DEMB;
        const unsigned blocks = (unsigned)((nz + 255) / 256);
        zero_kernel<<<blocks, 256, 0, stream>>>(agg, nz);
    }

    // 2) h = relu(x @ W_in + b_in)
    {
        const unsigned blocks = (unsigned)((N + 127) / 128);
        gemm128_relu_kernel<true, false><<<blocks, 256, 0, stream>>>(x, W_in, b_in, h, N);
    }

    // 3) agg[src] += edge_val * h[dst]
    {
        const unsigned blocks = (unsigned)(((long long)E * 32 + 255) / 256);
        spmm_edge_kernel<<<blocks, 256, 0, stream>>>(edge_src, edge_dst, edge_val, h, agg, E);
    }

    // 4) h_struct = l2norm(relu(agg @ W_gcn))
    {
        const unsigned blocks = (unsigned)((N + 127) / 128);
        gemm128_relu_kernel<false, true><<<blocks, 256, 0, stream>>>(agg, W_gcn, nullptr,
                                                                     h_struct, N);
    }

    // 5) pool[g] += batch_val[i] * h_struct[i]
    {
        const unsigned blocks = (unsigned)(((long long)N * 32 + 255) / 256);
        pool_node_kernel<<<blocks, 256, 0, stream>>>(h_struct, batch_vl, batch_id, pool, N);
    }

    // 6) h_graph = l2norm(relu(pool @ W_g + b_g))
    {
        const unsigned blocks = (unsigned)((G + 127) / 128);
        gemm128_relu_kernel<true, true><<<blocks, 256, 0, stream>>>(pool, W_g, b_g, h_graph, G);
    }
}